// GatedJunction_61272003445014
// MI455X (gfx1250) — compile-verified
//
#include <hip/hip_runtime.h>
#include <hip/hip_bf16.h>

typedef __attribute__((ext_vector_type(2))) float v2f;
typedef __attribute__((ext_vector_type(8))) float v8f;

#define Bsz 32
#define Kb  4
#define Cc  256
#define HWs 1024            // H*W
#define NF  1280            // (K+1)*C
#define MIDs 32
#define EPSc 1e-5f

// ---------------------------------------------------------------------------
// Kernel A: per-(b, branch, c) spatial mean. One wave32 per channel.
// 1024 floats / 32 lanes = 8 float4 loads per lane, then xor-shuffle reduce.
// ---------------------------------------------------------------------------
__global__ __launch_bounds__(256) void gj_reduce_means(
    const float* __restrict__ y,
    const float* __restrict__ x0, const float* __restrict__ x1,
    const float* __restrict__ x2, const float* __restrict__ x3,
    float* __restrict__ feats)           // [B][NF]
{
  const int lane = threadIdx.x & 31;
  const int wid  = threadIdx.x >> 5;
  const int ch   = blockIdx.x * 8 + wid;        // [0, B*NF)
  const int b = ch / NF;
  const int r = ch % NF;

  const float* src;
  if (r < Cc) {
    src = y + (size_t)(b * Cc + r) * HWs;
  } else {
    const int k = (r - Cc) >> 8;                // branch index
    const int c = (r - Cc) & 255;
    const float* xk = (k == 0) ? x0 : (k == 1) ? x1 : (k == 2) ? x2 : x3;
    src = xk + (size_t)(b * Cc + c) * HWs;
  }

  const float4* s4 = (const float4*)src;
  float s = 0.f;
#pragma unroll
  for (int i = 0; i < 8; ++i) {
    float4 v = s4[i * 32 + lane];
    s += (v.x + v.y) + (v.z + v.w);
  }
#pragma unroll
  for (int off = 16; off > 0; off >>= 1)
    s += __shfl_xor(s, off, 32);
  if (lane == 0) feats[ch] = s * (1.0f / 1024.0f);
}

// ---------------------------------------------------------------------------
// Kernel B: gate MLP with V_WMMA_F32_16X16X4_F32, one workgroup (4 waves).
//   h  = feats @ conv1_w^T          [32 x 32]
//   h  = relu(bn(h))
//   gw = h @ conv2_w^T + conv2_b    [32 x 1280]   (raw pre-activation gates)
// f32 WMMA layouts (ISA 7.12.2):
//   A 16x4 : lane(l16, hi): {A[l16][2*hi], A[l16][2*hi+1]}
//   B 4x16 : lane(l16, hi): {B[2*hi][l16], B[2*hi+1][l16]}
//   D 16x16: VGPR i -> D[8*hi + i][l16]
// ---------------------------------------------------------------------------
__global__ __launch_bounds__(128) void gj_gates_gemm(
    const float* __restrict__ feats,     // [B][NF]
    const float* __restrict__ conv1_w,   // [MID][NF]
    const float* __restrict__ bn_gamma, const float* __restrict__ bn_beta,
    const float* __restrict__ bn_mean,  const float* __restrict__ bn_var,
    const float* __restrict__ conv2_w,   // [NF][MID]
    const float* __restrict__ conv2_b,   // [NF]
    float* __restrict__ gw)              // [B][NF]
{
  __shared__ float s_h[Bsz][MIDs];

  const int lane = threadIdx.x & 31;
  const int wid  = threadIdx.x >> 5;    // 0..3
  const int l16  = lane & 15;
  const int hi   = lane >> 4;           // 0/1

  // ---- GEMM1: each wave owns one 16x16 tile of h ----
  {
    const int bt = wid >> 1;            // batch tile
    const int nt = wid & 1;             // MID tile
    const float* arow = feats   + (size_t)(bt * 16 + l16) * NF + 2 * hi;
    const float* brow = conv1_w + (size_t)(nt * 16 + l16) * NF + 2 * hi;
    v8f acc = {};
    for (int kk = 0; kk < NF; kk += 4) {
      v2f a;  a.x = arow[kk];  a.y = arow[kk + 1];
      v2f bb; bb.x = brow[kk]; bb.y = brow[kk + 1];
      acc = __builtin_amdgcn_wmma_f32_16x16x4_f32(
          false, a, false, bb, (short)0, acc, false, false);
    }
#pragma unroll
    for (int i = 0; i < 8; ++i)
      s_h[bt * 16 + 8 * hi + i][nt * 16 + l16] = acc[i];
  }
  __syncthreads();

  // ---- BN (inference) + ReLU on s_h: 1024 vals / 128 threads ----
#pragma unroll
  for (int i = 0; i < 8; ++i) {
    const int idx = threadIdx.x + i * 128;
    const int row = idx >> 5, col = idx & 31;
    float v = s_h[row][col];
    const float g = bn_gamma[col] * rsqrtf(bn_var[col] + EPSc);
    v = (v - bn_mean[col]) * g + bn_beta[col];
    s_h[row][col] = v > 0.f ? v : 0.f;
  }
  __syncthreads();

  // ---- GEMM2: 2 x 80 = 160 tiles, round-robin over 4 waves ----
  for (int t = wid; t < 160; t += 4) {
    const int bt = t & 1;
    const int nt = t >> 1;              // 0..79
    v8f acc = {};
#pragma unroll
    for (int kk = 0; kk < MIDs; kk += 4) {
      v2f a;
      a.x = s_h[bt * 16 + l16][kk + 2 * hi];
      a.y = s_h[bt * 16 + l16][kk + 2 * hi + 1];
      const float* brow = conv2_w + (size_t)(nt * 16 + l16) * MIDs + kk + 2 * hi;
      v2f bb; bb.x = brow[0]; bb.y = brow[1];
      acc = __builtin_amdgcn_wmma_f32_16x16x4_f32(
          false, a, false, bb, (short)0, acc, false, false);
    }
    const int col = nt * 16 + l16;
    const float bias = conv2_b[col];
#pragma unroll
    for (int i = 0; i < 8; ++i)
      gw[(size_t)(bt * 16 + 8 * hi + i) * NF + col] = acc[i] + bias;
  }
}

// ---------------------------------------------------------------------------
// Kernel C: gated combine. One block per (b,c) plane (1024 pixels = 256
// float4s). Gate scalars are block-uniform -> scalar loads + a few v_exp.
// out = y * sigmoid(w[b,0,c]) + sum_k softmax_k(w[b,1+k,c]) * xk
// ---------------------------------------------------------------------------
__global__ __launch_bounds__(256) void gj_combine(
    const float* __restrict__ y,
    const float* __restrict__ x0, const float* __restrict__ x1,
    const float* __restrict__ x2, const float* __restrict__ x3,
    const float* __restrict__ gw,        // [B][NF]
    float* __restrict__ out)
{
  const int bc = blockIdx.x;            // b*C + c
  const int b  = bc >> 8;
  const int c  = bc & 255;

  const float* g = gw + (size_t)b * NF;
  const float selfg = 1.0f / (1.0f + __expf(-g[c]));
  const float e0 = g[1 * Cc + c], e1 = g[2 * Cc + c];
  const float e2 = g[3 * Cc + c], e3 = g[4 * Cc + c];
  const float mx = fmaxf(fmaxf(e0, e1), fmaxf(e2, e3));
  float p0 = __expf(e0 - mx), p1 = __expf(e1 - mx);
  float p2 = __expf(e2 - mx), p3 = __expf(e3 - mx);
  const float inv = 1.0f / (p0 + p1 + p2 + p3);
  p0 *= inv; p1 *= inv; p2 *= inv; p3 *= inv;

  const size_t base = (size_t)bc * HWs;
  const float4* y4 = (const float4*)(y + base);
  const float4* a4 = (const float4*)(x0 + base);
  const float4* b4 = (const float4*)(x1 + base);
  const float4* c4 = (const float4*)(x2 + base);
  const float4* d4 = (const float4*)(x3 + base);
  float4* o4 = (float4*)(out + base);

  const int i = threadIdx.x;            // 0..255 covers 1024 floats
  const float4 vy = y4[i];
  const float4 v0 = a4[i];
  const float4 v1 = b4[i];
  const float4 v2 = c4[i];
  const float4 v3 = d4[i];
  float4 o;
  o.x = vy.x * selfg + p0 * v0.x + p1 * v1.x + p2 * v2.x + p3 * v3.x;
  o.y = vy.y * selfg + p0 * v0.y + p1 * v1.y + p2 * v2.y + p3 * v3.y;
  o.z = vy.z * selfg + p0 * v0.z + p1 * v1.z + p2 * v2.z + p3 * v3.z;
  o.w = vy.w * selfg + p0 * v0.w + p1 * v1.w + p2 * v2.w + p3 * v3.w;
  o4[i] = o;
}

// ---------------------------------------------------------------------------
extern "C" void kernel_launch(void* const* d_in, const int* in_sizes, int n_in,
                              void* d_out, int out_size, void* d_ws, size_t ws_size,
                              hipStream_t stream) {
  const float* y        = (const float*)d_in[0];
  const float* x0       = (const float*)d_in[1];
  const float* x1       = (const float*)d_in[2];
  const float* x2       = (const float*)d_in[3];
  const float* x3       = (const float*)d_in[4];
  const float* conv1_w  = (const float*)d_in[5];
  const float* bn_gamma = (const float*)d_in[6];
  const float* bn_beta  = (const float*)d_in[7];
  const float* bn_mean  = (const float*)d_in[8];
  const float* bn_var   = (const float*)d_in[9];
  const float* conv2_w  = (const float*)d_in[10];
  const float* conv2_b  = (const float*)d_in[11];
  float* out = (float*)d_out;

  float* feats = (float*)d_ws;          // [32][1280] = 160 KB
  float* gw    = feats + Bsz * NF;      // [32][1280] = 160 KB

  // Pass 1: 40960 channel means, 8 waves per block.
  gj_reduce_means<<<(Bsz * NF) / 8, 256, 0, stream>>>(y, x0, x1, x2, x3, feats);

  // Pass 2: tiny gate MLP on one WGP, f32 WMMA.
  gj_gates_gemm<<<1, 128, 0, stream>>>(feats, conv1_w, bn_gamma, bn_beta,
                                       bn_mean, bn_var, conv2_w, conv2_b, gw);

  // Pass 3: gated weighted sum, one block per (b,c) plane.
  gj_combine<<<Bsz * Cc, 256, 0, stream>>>(y, x0, x1, x2, x3, gw, out);
}